// RKGCN_40355512713612
// MI455X (gfx1250) — compile-verified
//
#include <hip/hip_runtime.h>
#include <math.h>

#define N_ENT 100000
#define N_RELS 16
#define DIM   32
#define N_HOP 2
#define N_MEM 32
#define N_NBR 16
#define BATCH 4096
#define TILE  16    // batch items per workgroup (WMMA M dimension)
#define NWAVE 4     // waves per workgroup
#define IPW   (TILE / NWAVE)   // items per wave in scalar phases

typedef float v2f __attribute__((ext_vector_type(2)));
typedef float v8f __attribute__((ext_vector_type(8)));

__device__ __forceinline__ float wsum(float v) {
#pragma unroll
  for (int m = 16; m >= 1; m >>= 1) v += __shfl_xor(v, m, 32);
  return v;
}
__device__ __forceinline__ float wmaxr(float v) {
#pragma unroll
  for (int m = 16; m >= 1; m >>= 1) v = fmaxf(v, __shfl_xor(v, m, 32));
  return v;
}

__global__ void rkgcn_init(float* out) { out[BATCH] = 0.0f; }

__global__ __launch_bounds__(NWAVE * 32)
void rkgcn_kernel(const int* __restrict__ items,
                  const int* __restrict__ mem_h,
                  const int* __restrict__ mem_r,
                  const int* __restrict__ mem_t,
                  const int* __restrict__ nbr_e,
                  const int* __restrict__ nbr_r,
                  const float* __restrict__ ent,
                  const float* __restrict__ relt,
                  const float* __restrict__ relg,
                  const float* __restrict__ tW,
                  const float* __restrict__ tb,
                  const float* __restrict__ gW,
                  const float* __restrict__ gb,
                  float* __restrict__ out)
{
  __shared__ float V [TILE][DIM + 1];            // current user embedding v
  __shared__ float U [TILE][N_RELS * DIM];       // U[b][rel*32+j] = (v_b^T R_rel)_j
  __shared__ float Hs[NWAVE][N_MEM][DIM + 1];    // per-wave h-row cache (KGE reuse)
  __shared__ float RM[N_RELS][DIM + 1];          // r_mean per relation
  __shared__ float X [TILE][DIM + 1];            // GEMM input staging
  __shared__ float CUR[TILE][DIM + 1];           // GCN entity state

  const int tid  = threadIdx.x;
  const int lane = tid & 31;
  const int wave = tid >> 5;
  const int half = lane >> 4;
  const int m16  = lane & 15;
  const int b0   = blockIdx.x * TILE;

  // initial v and original entity embedding (coalesced 128B row loads, items
  // distributed across waves)
  for (int it = wave; it < TILE; it += NWAVE) {
    int e = items[b0 + it];
    float val = ent[e * DIM + lane];
    V[it][lane]   = val;
    CUR[it][lane] = val;
  }

  // r_mean[rel][i] = mean_j R_rel[i][j]  (relation-only -> hoisted out of hops)
  for (int idx = tid; idx < N_RELS * DIM; idx += NWAVE * 32) {
    int r = idx >> 5, i = idx & 31;
    const float* p = relt + r * DIM * DIM + i * DIM;
    float s = 0.0f;
#pragma unroll
    for (int j = 0; j < DIM; ++j) s += p[j];
    RM[r][i] = s * (1.0f / DIM);
  }
  __syncthreads();

  float kge_acc = 0.0f;

  for (int hop = 0; hop < N_HOP; ++hop) {
    // ---- U(16x512) = V(16x32) @ Rperm(32x512), n-tiles split across waves ----
    for (int nt = wave; nt < (N_RELS * DIM) / 16; nt += NWAVE) {
      v8f acc = {};
#pragma unroll
      for (int kk = 0; kk < DIM / 4; ++kk) {             // 8 k-steps of K=4
        int kbase = kk * 4 + 2 * half;
        v2f a, bf;
        a.x = V[m16][kbase];
        a.y = V[m16][kbase + 1];
        int col = nt * 16 + m16;            // col = rel*32 + j
        int rId = col >> 5, j = col & 31;
        const float* bp = relt + rId * DIM * DIM + j;    // B[k][col] = R_rel[k][j]
        bf.x = bp[(kbase    ) * DIM];
        bf.y = bp[(kbase + 1) * DIM];
        acc = __builtin_amdgcn_wmma_f32_16x16x4_f32(false, a, false, bf,
                                                    (short)0, acc, false, false);
      }
#pragma unroll
      for (int g = 0; g < 8; ++g)
        U[g + 8 * half][nt * 16 + m16] = acc[g];
    }
    __syncthreads();

    const int hofs = hop * BATCH * N_MEM;

    // attention: each wave owns IPW items (4x less serial latency per wave)
    for (int k = 0; k < IPW; ++k) {
      const int it = wave * IPW + k;
      const int b  = b0 + it;
      // phase A: att_logit[n] = (v^T R_{rel(n)}) . h_n   (lane = feature j)
      float logit = 0.0f;
      for (int n = 0; n < N_MEM; ++n) {
        int hi = mem_h[hofs + b * N_MEM + n];
        int rr = mem_r[hofs + b * N_MEM + n];
        float hj = ent[hi * DIM + lane];
        Hs[wave][n][lane] = hj;
        float red = wsum(hj * U[it][rr * DIM + lane]);
        if (lane == n) logit = red;
      }
      // softmax over 32 memories (one per lane)
      float mx  = wmaxr(logit);
      float e   = __expf(logit - mx);
      float s   = wsum(e);
      float att = e / s;
      // phase B: o_j = sum_n att_n * t_nj ; fused KGE residual
      float o = 0.0f;
      for (int n = 0; n < N_MEM; ++n) {
        int ti = mem_t[hofs + b * N_MEM + n];
        int rr = mem_r[hofs + b * N_MEM + n];
        float tj = ent[ti * DIM + lane];
        float a  = __shfl(att, n, 32);
        o += a * tj;
        float diff = Hs[wave][n][lane] + RM[rr][lane] - tj;
        float red = wsum(diff * diff);
        if (lane == 0) kge_acc += red;
      }
      X[it][lane] = o + V[it][lane];
    }
    __syncthreads();

    // ---- v = tanh(X @ tW[hop] + tb[hop]) : 2 n-tiles on waves 0..1 ----
    if (wave < DIM / 16) {
      const float* W = tW + hop * DIM * DIM;
      const int nt = wave;
      v8f acc = {};
#pragma unroll
      for (int kk = 0; kk < DIM / 4; ++kk) {
        int kbase = kk * 4 + 2 * half;
        v2f a, bf;
        a.x = X[m16][kbase];
        a.y = X[m16][kbase + 1];
        int j = nt * 16 + m16;
        bf.x = W[(kbase    ) * DIM + j];
        bf.y = W[(kbase + 1) * DIM + j];
        acc = __builtin_amdgcn_wmma_f32_16x16x4_f32(false, a, false, bf,
                                                    (short)0, acc, false, false);
      }
#pragma unroll
      for (int g = 0; g < 8; ++g) {
        int row = g + 8 * half;
        int j = nt * 16 + m16;
        V[row][j] = tanhf(acc[g] + tb[hop * DIM + j]);
      }
    }
    __syncthreads();
  }

  // ---- GCN layer: each wave owns IPW items ----
  for (int k = 0; k < IPW; ++k) {
    const int it = wave * IPW + k;
    const int b  = b0 + it;
    int base = items[b] * N_NBR;
    float logit = -1e30f;
    for (int n = 0; n < N_NBR; ++n) {
      int rr = nbr_r[base + n];
      float red = wsum(V[it][lane] * relg[rr * DIM + lane]);
      if (lane == n) logit = red;
    }
    float mx  = wmaxr(logit);
    float e   = (lane < N_NBR) ? __expf(logit - mx) : 0.0f;
    float s   = wsum(e);
    float att = e / s;
    float agg = 0.0f;
    for (int n = 0; n < N_NBR; ++n) {
      int ei = nbr_e[base + n];
      float a = __shfl(att, n, 32);
      agg += a * ent[ei * DIM + lane];
    }
    X[it][lane] = CUR[it][lane] + agg;
  }
  __syncthreads();

  // cur = relu(X @ gW + gb) : 2 n-tiles on waves 0..1
  if (wave < DIM / 16) {
    const int nt = wave;
    v8f acc = {};
#pragma unroll
    for (int kk = 0; kk < DIM / 4; ++kk) {
      int kbase = kk * 4 + 2 * half;
      v2f a, bf;
      a.x = X[m16][kbase];
      a.y = X[m16][kbase + 1];
      int j = nt * 16 + m16;
      bf.x = gW[(kbase    ) * DIM + j];
      bf.y = gW[(kbase + 1) * DIM + j];
      acc = __builtin_amdgcn_wmma_f32_16x16x4_f32(false, a, false, bf,
                                                  (short)0, acc, false, false);
    }
#pragma unroll
    for (int g = 0; g < 8; ++g) {
      int row = g + 8 * half;
      int j = nt * 16 + m16;
      CUR[row][j] = fmaxf(acc[g] + gb[j], 0.0f);
    }
  }
  __syncthreads();

  // logits + sigmoid (items distributed across waves)
  for (int it = wave; it < TILE; it += NWAVE) {
    float red = wsum(V[it][lane] * CUR[it][lane]);
    if (lane == 0) out[b0 + it] = 1.0f / (1.0f + __expf(-red));
  }

  // kge_loss: both hops use mean over (BATCH*N_MEM); one atomic per wave
  if (lane == 0)
    atomicAdd(out + BATCH, kge_acc * (1.0f / ((float)BATCH * (float)N_MEM)));
}

extern "C" void kernel_launch(void* const* d_in, const int* in_sizes, int n_in,
                              void* d_out, int out_size, void* d_ws, size_t ws_size,
                              hipStream_t stream) {
  const int*   items = (const int*)  d_in[0];
  const int*   mem_h = (const int*)  d_in[1];
  const int*   mem_r = (const int*)  d_in[2];
  const int*   mem_t = (const int*)  d_in[3];
  const int*   nbr_e = (const int*)  d_in[4];
  const int*   nbr_r = (const int*)  d_in[5];
  const float* ent   = (const float*)d_in[6];
  const float* relt  = (const float*)d_in[7];
  const float* relg  = (const float*)d_in[8];
  const float* tW    = (const float*)d_in[9];
  const float* tb    = (const float*)d_in[10];
  const float* gW    = (const float*)d_in[11];
  const float* gb    = (const float*)d_in[12];
  float* out = (float*)d_out;

  rkgcn_init<<<1, 1, 0, stream>>>(out);
  rkgcn_kernel<<<BATCH / TILE, NWAVE * 32, 0, stream>>>(
      items, mem_h, mem_r, mem_t, nbr_e, nbr_r,
      ent, relt, relg, tW, tb, gW, gb, out);
}